// DynamicHead_86260123174144
// MI455X (gfx1250) — compile-verified
//
#include <hip/hip_runtime.h>
#include <math.h>

// CDNA5 / gfx1250, wave32.
// Fused DynamicHead addressing kernel.
// Memory-bound: memory_vb (256 MB) dominates -> single pass, everything fused.
// GEMM hidden@W_key done with V_WMMA_F32_16X16X4_F32 (full f32 precision; compute
// is ~2.7 GFLOP total, far below the ~12.5us HBM floor at 23.3 TB/s).
// tanh uses native V_TANH_F32 when the builtin is available (gfx1250 TRANS op),
// avoiding the branchy libm expansion.

typedef float v2f __attribute__((ext_vector_type(2)));
typedef float v8f __attribute__((ext_vector_type(8)));

#if __has_builtin(__builtin_amdgcn_tanhf)
__device__ __forceinline__ float fast_tanh(float x) { return __builtin_amdgcn_tanhf(x); }
#else
__device__ __forceinline__ float fast_tanh(float x) { return tanhf(x); }
#endif

#define D_DIM  512
#define HW_DIM 256
#define H_DIM  4
#define M_DIM  128
#define W_DIM  64
#define ROWS   16          // batch rows per block
#define HS     516         // hidden_s row stride (pad: 516%64==4 -> conflict-free b64 A reads)
#define KS     260         // key_s row stride (16B-aligned rows)
#define EPSC   1e-6f

__global__ __launch_bounds__(256)
void dynamic_head_fused(const float* __restrict__ hidden,
                        const float* __restrict__ mem,
                        const float* __restrict__ Wk,
                        const float* __restrict__ bk,
                        const float* __restrict__ Wb,
                        const float* __restrict__ bb,
                        float* __restrict__ out)
{
    __shared__ float hidden_s[ROWS * HS];     // 33.0 KB
    __shared__ float key_s[ROWS * KS];        // 16.6 KB
    __shared__ float s_s[8 * M_DIM];          // 4 KB  [bl(2)][h(4)][m(128)]
    __shared__ float beta_s[ROWS * H_DIM];
    __shared__ float u2_s[ROWS * H_DIM];

    const int t    = threadIdx.x;
    const int lane = t & 31;
    const int wv   = t >> 5;                   // wave id 0..7
    const int base = blockIdx.x * ROWS;

    // ---------------- Phase 1a: stage hidden tile (coalesced b128) ----------------
    {
        const float4* hid4 = (const float4*)hidden;
        #pragma unroll
        for (int it = 0; it < 8; ++it) {
            int s  = it * 256 + t;             // 2048 float4 slots
            int r  = s >> 7;                   // 128 float4 per row
            int c4 = s & 127;
            float4 g = hid4[(size_t)(base + r) * (D_DIM / 4) + c4];
            *(float4*)&hidden_s[r * HS + c4 * 4] = g;
        }
    }
    __syncthreads();

    // ------------- Phase 1b: key = tanh(hidden @ Wk + bk), WMMA f32 16x16x4 -------
    // A 16x4 layout: VGPR0 = K {0 | 2}, VGPR1 = K {1 | 3} across lane halves (m = lane&15).
    // B 4x16 layout: VGPR0 = row K {0 | 2}, VGPR1 = row K {1 | 3}, n = lane&15.
    // C 16x16: acc[r] -> M = r + (lane>=16 ? 8 : 0), N = lane&15.
    {
        const int m16  = lane & 15;
        const int koff = (lane >= 16) ? 2 : 0;
        const int n0   = wv * 16 + m16;        // wave owns N-tiles wv and wv+8
        const int n1   = (wv + 8) * 16 + m16;
        v8f acc0 = {0.f,0.f,0.f,0.f,0.f,0.f,0.f,0.f};
        v8f acc1 = {0.f,0.f,0.f,0.f,0.f,0.f,0.f,0.f};
        for (int k = 0; k < D_DIM; k += 4) {
            const int kk = k + koff;
            v2f a = *(const v2f*)&hidden_s[m16 * HS + kk];   // 8B-aligned, conflict-free
            v2f b0, b1;
            b0.x = Wk[kk * HW_DIM + n0];  b0.y = Wk[(kk + 1) * HW_DIM + n0];
            b1.x = Wk[kk * HW_DIM + n1];  b1.y = Wk[(kk + 1) * HW_DIM + n1];
            acc0 = __builtin_amdgcn_wmma_f32_16x16x4_f32(false, a, false, b0,
                                                         (short)0, acc0, false, false);
            acc1 = __builtin_amdgcn_wmma_f32_16x16x4_f32(false, a, false, b1,
                                                         (short)0, acc1, false, false);
        }
        const float bias0 = bk[n0], bias1 = bk[n1];
        const int mb = (lane >= 16) ? 8 : 0;
        #pragma unroll
        for (int r = 0; r < 8; ++r) {
            key_s[(mb + r) * KS + n0] = fast_tanh(acc0[r] + bias0);
            key_s[(mb + r) * KS + n1] = fast_tanh(acc1[r] + bias1);
        }
    }
    __syncthreads();

    // ------------- Phase 1c: beta = softplus(hidden@Wb + bb), u2 = sum(key^2) -----
    {
        const int r = t >> 4, h = (t >> 2) & 3, q = t & 3;   // unique (r,h,q)
        float pb = 0.f, pu = 0.f;
        for (int i = 0; i < 128; ++i) {
            int d = q * 128 + i;
            pb += hidden_s[r * HS + d] * Wb[d * H_DIM + h];
        }
        #pragma unroll
        for (int i = 0; i < 16; ++i) {
            float kv = key_s[r * KS + h * W_DIM + q * 16 + i];
            pu += kv * kv;
        }
        pb += __shfl_xor(pb, 1, 32);  pb += __shfl_xor(pb, 2, 32);
        pu += __shfl_xor(pu, 1, 32);  pu += __shfl_xor(pu, 2, 32);
        if (q == 0) {
            float x = pb + bb[h];
            // numerically-stable softplus
            beta_s[r * H_DIM + h] = fmaxf(x, 0.f) + log1pf(__expf(-fabsf(x)));
            u2_s[r * H_DIM + h]   = pu + EPSC;
        }
    }
    __syncthreads();

    // ------------- Phase 2: stream memory_vb once; cosine-sim + softmax -----------
    const int bl = t >> 7;                     // 0..1  (2 batches per iteration)
    const int m  = t & 127;                    // M-row owned by this thread
    for (int ib = 0; ib < 8; ++ib) {
        const int rloc = ib * 2 + bl;          // local batch row 0..15
        const int b    = base + rloc;
        const float4* mrow = (const float4*)&mem[((size_t)b * M_DIM + m) * W_DIM];
        if (ib < 7)   // hint next pair of batches (emits global_prefetch_b8)
            __builtin_prefetch(&mem[((size_t)(b + 2) * M_DIM + m) * W_DIM], 0, 0);

        float vsq = EPSC;
        float n0 = 0.f, n1 = 0.f, n2 = 0.f, n3 = 0.f;
        const float* kr = &key_s[rloc * KS];
        #pragma unroll 4
        for (int j = 0; j < 16; ++j) {
            float4 mv = mrow[j];               // 256B/row contiguous per thread
            vsq += mv.x * mv.x + mv.y * mv.y + mv.z * mv.z + mv.w * mv.w;
            float4 k0 = *(const float4*)&kr[0 * W_DIM + j * 4];   // LDS broadcast
            float4 k1 = *(const float4*)&kr[1 * W_DIM + j * 4];
            float4 k2 = *(const float4*)&kr[2 * W_DIM + j * 4];
            float4 k3 = *(const float4*)&kr[3 * W_DIM + j * 4];
            n0 += mv.x*k0.x + mv.y*k0.y + mv.z*k0.z + mv.w*k0.w;
            n1 += mv.x*k1.x + mv.y*k1.y + mv.z*k1.z + mv.w*k1.w;
            n2 += mv.x*k2.x + mv.y*k2.y + mv.z*k2.z + mv.w*k2.w;
            n3 += mv.x*k3.x + mv.y*k3.y + mv.z*k3.z + mv.w*k3.w;
        }
        // s = beta * num / (sqrt(u2e * v2e) + eps)  (fast div: well under roofline)
        {
            const float* u2p = &u2_s[rloc * H_DIM];
            const float* bep = &beta_s[rloc * H_DIM];
            s_s[(bl * 4 + 0) * M_DIM + m] = bep[0] * __fdividef(n0, sqrtf(u2p[0] * vsq) + EPSC);
            s_s[(bl * 4 + 1) * M_DIM + m] = bep[1] * __fdividef(n1, sqrtf(u2p[1] * vsq) + EPSC);
            s_s[(bl * 4 + 2) * M_DIM + m] = bep[2] * __fdividef(n2, sqrtf(u2p[2] * vsq) + EPSC);
            s_s[(bl * 4 + 3) * M_DIM + m] = bep[3] * __fdividef(n3, sqrtf(u2p[3] * vsq) + EPSC);
        }
        __syncthreads();

        // one wave per (batch, head): softmax over M=128 and coalesced store
        {
            const int pbl = wv >> 2, ph = wv & 3;
            const float* sp = &s_s[(pbl * 4 + ph) * M_DIM];
            float x0 = sp[lane], x1 = sp[lane + 32], x2 = sp[lane + 64], x3 = sp[lane + 96];
            float mx = fmaxf(fmaxf(x0, x1), fmaxf(x2, x3));
            #pragma unroll
            for (int msk = 16; msk >= 1; msk >>= 1)
                mx = fmaxf(mx, __shfl_xor(mx, msk, 32));
            float e0 = __expf(x0 - mx), e1 = __expf(x1 - mx);
            float e2 = __expf(x2 - mx), e3 = __expf(x3 - mx);
            float sm = e0 + e1 + e2 + e3;
            #pragma unroll
            for (int msk = 16; msk >= 1; msk >>= 1)
                sm += __shfl_xor(sm, msk, 32);
            float inv = __fdividef(1.f, sm);
            const int bg = base + ib * 2 + pbl;
            float* op = &out[((size_t)bg * H_DIM + ph) * M_DIM];
            op[lane]      = e0 * inv;
            op[lane + 32] = e1 * inv;
            op[lane + 64] = e2 * inv;
            op[lane + 96] = e3 * inv;
        }
        __syncthreads();   // protect s_s reuse next iteration
    }
}

extern "C" void kernel_launch(void* const* d_in, const int* in_sizes, int n_in,
                              void* d_out, int out_size, void* d_ws, size_t ws_size,
                              hipStream_t stream) {
    (void)n_in; (void)out_size; (void)d_ws; (void)ws_size;
    const float* hidden = (const float*)d_in[0];
    const float* mem    = (const float*)d_in[1];
    const float* Wk     = (const float*)d_in[2];
    const float* bk     = (const float*)d_in[3];
    const float* Wb     = (const float*)d_in[4];
    const float* bb     = (const float*)d_in[5];
    float* out = (float*)d_out;

    const int B = in_sizes[0] / D_DIM;         // 8192
    dim3 grid(B / ROWS), block(256);
    hipLaunchKernelGGL(dynamic_head_fused, grid, block, 0, stream,
                       hidden, mem, Wk, bk, Wb, bb, out);
}